// UltralyticsPose_65773129171254
// MI455X (gfx1250) — compile-verified
//
#include <hip/hip_runtime.h>
#include <hip/hip_bf16.h>
#include <math.h>

// ---------------------------------------------------------------------------
// YOLO-pose head for MI455X (gfx1250, wave32, WMMA).
// Stage 0: pack cv2/cv3/cv4 1x1-conv weights into one padded 208xC matrix.
// Stage 1: fused WMMA GEMM (fp32 16x16x4) + DFL/sigmoid/kpt decode per tile.
//          13 waves/block, each wave owns one 16-wide N tile of the 208
//          output channels for a shared 16-anchor M tile. Results staged in
//          LDS, decoded in-block; only compact per-anchor results hit HBM.
// Stage 2: per-image top-100 selection + gather.
// ---------------------------------------------------------------------------

typedef float v2f __attribute__((ext_vector_type(2)));
typedef float v8f __attribute__((ext_vector_type(8)));

#define NCH      208   // 64 box + 80 cls + 51 kpt, padded to 13*16
#define A_TOT    8400
#define BS       16
#define MAXDET   100
#define CONF_T   0.25f

__device__ __forceinline__ float sigmoidf_(float v) {
    return 1.0f / (1.0f + __expf(-v));
}

// ---------------- Stage 0: weight packing --------------------------------
__global__ void pose_pack_kernel(const float* __restrict__ w2, const float* __restrict__ b2,
                                 const float* __restrict__ w3, const float* __restrict__ b3,
                                 const float* __restrict__ w4, const float* __restrict__ b4,
                                 int C, float* __restrict__ Wp, float* __restrict__ biasp)
{
    int i = blockIdx.x * 256 + threadIdx.x;
    int total = NCH * C;
    if (i < total) {
        int row = i / C, c = i % C;
        float v = 0.0f;
        if (row < 64)        v = w2[row * C + c];
        else if (row < 144)  v = w3[(row - 64) * C + c];
        else if (row < 195)  v = w4[(row - 144) * C + c];
        Wp[i] = v;
    }
    if (i < NCH) {
        float v = 0.0f;
        if (i < 64)          v = b2[i];
        else if (i < 144)    v = b3[i - 64];
        else if (i < 195)    v = b4[i - 144];
        biasp[i] = v;
    }
}

// ---------------- Stage 1: fused WMMA GEMM + decode ----------------------
// grid: (HW/16, BS), block: 416 threads (13 waves)
__global__ __launch_bounds__(416)
void pose_head_kernel(const float* __restrict__ x, const float* __restrict__ Wp,
                      const float* __restrict__ biasp, int C, int HW, int Wdim,
                      float stride_v, int anchor_off,
                      float* __restrict__ S, float* __restrict__ B4,
                      float* __restrict__ CLS, float* __restrict__ KPT)
{
    __shared__ float tile[16][NCH + 1];   // [anchor][channel], +1 pad
    __shared__ float dists[16][4];

    const int b    = blockIdx.y;
    const int p0   = blockIdx.x * 16;
    const int t    = threadIdx.x;
    const int wave = t >> 5;
    const int lane = t & 31;
    const int half = lane >> 4;     // 0: K pair {0,1}, 1: K pair {2,3}
    const int mn   = lane & 15;     // M index for A, N index for B/D
    const int n0   = wave * 16;
    const int klo  = half * 2;

    // A: 16x4 fp32 frag. lane<16 -> K=0,1 ; lane>=16 -> K=2,3 (per-VGPR K stripe)
    const float* __restrict__ xb   = x + (size_t)b * C * HW + p0 + mn;
    // B: 4x16 fp32 frag. lane<16 -> rows K=0,1 ; lane>=16 -> rows K=2,3
    const float* __restrict__ wrow = Wp + (size_t)(n0 + mn) * C;

    v8f acc = {0.f, 0.f, 0.f, 0.f, 0.f, 0.f, 0.f, 0.f};
    #pragma unroll 4
    for (int k0 = 0; k0 < C; k0 += 4) {
        v2f a, bb;
        a.x = xb[(size_t)(k0 + klo)     * HW];
        a.y = xb[(size_t)(k0 + klo + 1) * HW];
        bb  = *(const v2f*)(wrow + k0 + klo);
        acc = __builtin_amdgcn_wmma_f32_16x16x4_f32(
                  false, a, false, bb, (short)0, acc, false, false);
    }

    // D layout: vgpr j -> M = j + 8*half, N = mn. Stage (+bias) to LDS.
    const float bias = biasp[n0 + mn];
    #pragma unroll
    for (int j = 0; j < 8; ++j)
        tile[j + half * 8][n0 + mn] = acc[j] + bias;
    __syncthreads();

    // ---------------- in-block decode of 16 anchors ----------------------
    if (t < 272) {                       // 16 anchors x 17 keypoints
        int a  = t / 17, k = t % 17;
        int la = p0 + a;
        float gxm = (float)(la % Wdim);  // gx - 0.5
        float gym = (float)(la / Wdim);  // gy - 0.5
        int cb = 144 + k * 3;
        float vx = tile[a][cb + 0], vy = tile[a][cb + 1], vv = tile[a][cb + 2];
        size_t ga = (size_t)b * A_TOT + anchor_off + la;
        float* kp = KPT + ga * 51 + (size_t)k * 3;
        kp[0] = (vx * 2.0f + gxm) * stride_v;
        kp[1] = (vy * 2.0f + gym) * stride_v;
        kp[2] = sigmoidf_(vv);
    } else if (t < 336) {                // 16 anchors x 4 DFL softmaxes
        int i = t - 272, a = i >> 2, f = i & 3;
        const float* lg = &tile[a][f * 16];
        float mx = lg[0];
        #pragma unroll
        for (int r = 1; r < 16; ++r) mx = fmaxf(mx, lg[r]);
        float se = 0.f, sw = 0.f;
        #pragma unroll
        for (int r = 0; r < 16; ++r) {
            float e = __expf(lg[r] - mx);
            se += e; sw += e * (float)r;
        }
        dists[a][f] = sw / se;
    } else if (t < 352) {                // 16 anchors: class max/argmax
        int a = t - 336, la = p0 + a;
        float mx = tile[a][64]; int id = 0;
        for (int c = 1; c < 80; ++c) {
            float v = tile[a][64 + c];
            if (v > mx) { mx = v; id = c; }
        }
        size_t ga = (size_t)b * A_TOT + anchor_off + la;
        S[ga]   = sigmoidf_(mx);         // sigmoid is monotonic
        CLS[ga] = (float)id;
    }
    __syncthreads();

    if (t < 16) {                        // boxes from DFL distances
        int a = t, la = p0 + a;
        float gx = (float)(la % Wdim) + 0.5f;
        float gy = (float)(la / Wdim) + 0.5f;
        float lt0 = dists[a][0], lt1 = dists[a][1];
        float rb0 = dists[a][2], rb1 = dists[a][3];
        size_t ga = (size_t)b * A_TOT + anchor_off + la;
        float* bx = B4 + ga * 4;
        bx[0] = (gx + 0.5f * (rb0 - lt0)) * stride_v;   // cx
        bx[1] = (gy + 0.5f * (rb1 - lt1)) * stride_v;   // cy
        bx[2] = (lt0 + rb0) * stride_v;                 // w
        bx[3] = (lt1 + rb1) * stride_v;                 // h
    }
}

// ---------------- Stage 2: per-image top-100 + gather --------------------
__global__ __launch_bounds__(1024)
void pose_topk_kernel(const float* __restrict__ S, const float* __restrict__ B4,
                      const float* __restrict__ CLS, const float* __restrict__ KPT,
                      float* __restrict__ out)
{
    __shared__ float sc[A_TOT];
    __shared__ float rv[1024];
    __shared__ int   ri[1024];
    __shared__ int   topi[MAXDET];
    __shared__ float tops[MAXDET];

    const int b = blockIdx.x;
    const int t = threadIdx.x;
    const float* Sb = S + (size_t)b * A_TOT;
    for (int i = t; i < A_TOT; i += 1024) sc[i] = Sb[i];
    __syncthreads();

    for (int r = 0; r < MAXDET; ++r) {
        float bv = -1.0f; int bi = 0;
        for (int i = t; i < A_TOT; i += 1024) {
            float v = sc[i];
            if (v > bv) { bv = v; bi = i; }
        }
        rv[t] = bv; ri[t] = bi;
        __syncthreads();
        for (int s = 512; s > 0; s >>= 1) {
            if (t < s) {
                float ov = rv[t + s]; int oi = ri[t + s];
                if (ov > rv[t] || (ov == rv[t] && oi < ri[t])) { rv[t] = ov; ri[t] = oi; }
            }
            __syncthreads();
        }
        if (t == 0) { topi[r] = ri[0]; tops[r] = rv[0]; sc[ri[0]] = -1.0f; }
        __syncthreads();
    }

    // output layout (flat, return order): num_dets | det_boxes | top_s | det_classes | det_kpts
    float* o_nd  = out;                               // 16*1
    float* o_box = out + BS;                          // 16*100*4
    float* o_s   = o_box + (size_t)BS * MAXDET * 4;   // 16*100
    float* o_cls = o_s   + (size_t)BS * MAXDET;       // 16*100
    float* o_kpt = o_cls + (size_t)BS * MAXDET;       // 16*100*17*3

    if (t == 0) {
        int c = 0;
        for (int r = 0; r < MAXDET; ++r) if (tops[r] > CONF_T) ++c;
        o_nd[b] = (float)c;
    }
    for (int i = t; i < MAXDET; i += 1024) o_s[(size_t)b * MAXDET + i] = tops[i];
    for (int idx = t; idx < MAXDET * 56; idx += 1024) {
        int r = idx / 56, c = idx % 56;
        size_t ga = (size_t)b * A_TOT + topi[r];
        if (c < 4)       o_box[((size_t)b * MAXDET + r) * 4 + c] = B4[ga * 4 + c];
        else if (c == 4) o_cls[(size_t)b * MAXDET + r]           = CLS[ga];
        else             o_kpt[((size_t)b * MAXDET + r) * 51 + (c - 5)] = KPT[ga * 51 + (c - 5)];
    }
}

// ---------------------------------------------------------------------------
extern "C" void kernel_launch(void* const* d_in, const int* in_sizes, int n_in,
                              void* d_out, int out_size, void* d_ws, size_t ws_size,
                              hipStream_t stream) {
    (void)in_sizes; (void)n_in; (void)out_size; (void)ws_size;
    const float* x0 = (const float*)d_in[0];
    const float* x1 = (const float*)d_in[1];
    const float* x2 = (const float*)d_in[2];
    const float* cv2_w[3] = { (const float*)d_in[3], (const float*)d_in[5], (const float*)d_in[7] };
    const float* cv2_b[3] = { (const float*)d_in[4], (const float*)d_in[6], (const float*)d_in[8] };
    const float* cv3_w[3] = { (const float*)d_in[9],  (const float*)d_in[11], (const float*)d_in[13] };
    const float* cv3_b[3] = { (const float*)d_in[10], (const float*)d_in[12], (const float*)d_in[14] };
    const float* cv4_w[3] = { (const float*)d_in[15], (const float*)d_in[17], (const float*)d_in[19] };
    const float* cv4_b[3] = { (const float*)d_in[16], (const float*)d_in[18], (const float*)d_in[20] };

    const int   Cs[3]   = { 128, 256, 512 };
    const int   Ws[3]   = { 80, 40, 20 };
    const float Sv[3]   = { 8.0f, 16.0f, 32.0f };
    const int   Aoff[3] = { 0, 6400, 8000 };
    const float* xs[3]  = { x0, x1, x2 };

    // scratch layout (floats)
    float* ws = (float*)d_ws;
    float* Wp[3];
    Wp[0] = ws;                              // 208*128
    Wp[1] = Wp[0] + NCH * 128;               // 208*256
    Wp[2] = Wp[1] + NCH * 256;               // 208*512
    float* biasp = Wp[2] + NCH * 512;        // 3*208
    float* S     = biasp + 3 * NCH;          // 16*8400
    float* CLS   = S   + (size_t)BS * A_TOT; // 16*8400
    float* B4    = CLS + (size_t)BS * A_TOT; // 16*8400*4
    float* KPT   = B4  + (size_t)BS * A_TOT * 4; // 16*8400*51

    for (int s = 0; s < 3; ++s) {
        int C = Cs[s];
        int n = NCH * C;
        pose_pack_kernel<<<(n + 255) / 256, 256, 0, stream>>>(
            cv2_w[s], cv2_b[s], cv3_w[s], cv3_b[s], cv4_w[s], cv4_b[s],
            C, Wp[s], biasp + s * NCH);
    }
    for (int s = 0; s < 3; ++s) {
        int C = Cs[s], Wd = Ws[s], HW = Wd * Wd;
        dim3 grid(HW / 16, BS);
        pose_head_kernel<<<grid, 416, 0, stream>>>(
            xs[s], Wp[s], biasp + s * NCH, C, HW, Wd, Sv[s], Aoff[s],
            S, B4, CLS, KPT);
    }
    pose_topk_kernel<<<BS, 1024, 0, stream>>>(S, B4, CLS, KPT, (float*)d_out);
}